// _KLD_MVNx2_8744553415074
// MI455X (gfx1250) — compile-verified
//
#include <hip/hip_runtime.h>
#include <hip/hip_bf16.h>

typedef __attribute__((ext_vector_type(2))) float v2f;
typedef __attribute__((ext_vector_type(4))) float v4f;
typedef __attribute__((ext_vector_type(8))) float v8f;

#define KL_NB    1024   // partial blocks
#define KL_BLOCK 256    // 8 waves (wave32)

// Per-element KL contribution:
//   t = 0.5 * ( (sq - sp) + exp(sp - sq) + (mq-mp)^2 * exp(-sq) - 1 )
__device__ __forceinline__ float kl_elem(float mq, float sq, float mp, float sp) {
    float d = mq - mp;
    return 0.5f * ((sq - sp) + __expf(sp - sq) + d * d * __expf(-sq) - 1.0f);
}

__global__ __launch_bounds__(KL_BLOCK) void _KLD_partials_kernel(
    const float* __restrict__ mu_q, const float* __restrict__ sigma_q,
    const float* __restrict__ mu_p, const float* __restrict__ sigma_p,
    float* __restrict__ partials, int n)
{
    const int tid    = blockIdx.x * blockDim.x + threadIdx.x;
    const int stride = gridDim.x * blockDim.x;
    const int n4     = n >> 2;

    const v4f* __restrict__ mq4 = (const v4f*)mu_q;
    const v4f* __restrict__ sq4 = (const v4f*)sigma_q;
    const v4f* __restrict__ mp4 = (const v4f*)mu_p;
    const v4f* __restrict__ sp4 = (const v4f*)sigma_p;

    float acc = 0.0f;

    // Streaming phase: 4x b128 non-temporal loads per iteration (zero reuse,
    // 268MB total > 192MB L2 -> NT keeps L2 clean).
    for (int i = tid; i < n4; i += stride) {
        v4f mq = __builtin_nontemporal_load(&mq4[i]);
        v4f sq = __builtin_nontemporal_load(&sq4[i]);
        v4f mp = __builtin_nontemporal_load(&mp4[i]);
        v4f sp = __builtin_nontemporal_load(&sp4[i]);
        acc += kl_elem(mq.x, sq.x, mp.x, sp.x);
        acc += kl_elem(mq.y, sq.y, mp.y, sp.y);
        acc += kl_elem(mq.z, sq.z, mp.z, sp.z);
        acc += kl_elem(mq.w, sq.w, mp.w, sp.w);
    }
    // Scalar tail (n % 4), empty for N = 16M but kept for generality.
    for (int i = (n4 << 2) + tid; i < n; i += stride) {
        acc += kl_elem(mu_q[i], sigma_q[i], mu_p[i], sigma_p[i]);
    }

    // wave32 shuffle tree (deterministic order)
    #pragma unroll
    for (int off = 16; off > 0; off >>= 1)
        acc += __shfl_down(acc, off, 32);

    __shared__ float red[KL_BLOCK / 32];
    const int lane = threadIdx.x & 31;
    const int wid  = threadIdx.x >> 5;
    if (lane == 0) red[wid] = acc;
    __syncthreads();

    if (wid == 0) {
        float v = (lane < (KL_BLOCK / 32)) ? red[lane] : 0.0f;
        #pragma unroll
        for (int off = 4; off > 0; off >>= 1)   // 8 wave-partials -> 1
            v += __shfl_down(v, off, 32);
        if (lane == 0) partials[blockIdx.x] = v;
    }
}

// One wave (EXEC all 1s as WMMA requires). Each lane serially sums NB/32
// partials (deterministic), then V_WMMA_F32_16X16X4_F32 with A=ones,
// B=(x,0) collapses the 32 lane values into 16 column sums:
//   D[m][n] = sum_k B[k][n]  => sum_n D[0][n] = sum over ALL B slots = sum x
// (correct for any bijective lane->slot striping of B's first VGPR).
// Four xor-shuffles finish 16 -> 1.
__global__ __launch_bounds__(32) void _KLD_finalize_kernel(
    const float* __restrict__ partials, float* __restrict__ out, int nb)
{
    const int lane = threadIdx.x;
    float x = 0.0f;
    for (int i = lane; i < nb; i += 32)
        x += partials[i];

    v2f a; a.x = 1.0f; a.y = 1.0f;   // A = all ones (both K-striped VGPRs)
    v2f b; b.x = x;    b.y = 0.0f;   // lane partials in B VGPR0, VGPR1 = 0
    v8f c = {};
    v8f d = __builtin_amdgcn_wmma_f32_16x16x4_f32(
        /*neg_a=*/false, a, /*neg_b=*/false, b,
        /*c_mod=*/(short)0, c, /*reuse_a=*/false, /*reuse_b=*/false);

    float s = d[0];          // lane L holds colsum_{L mod 16}
    s += __shfl_xor(s, 1);
    s += __shfl_xor(s, 2);
    s += __shfl_xor(s, 4);
    s += __shfl_xor(s, 8);   // every lane now holds the grand total

    if (lane == 0) out[0] = s;
}

extern "C" void kernel_launch(void* const* d_in, const int* in_sizes, int n_in,
                              void* d_out, int out_size, void* d_ws, size_t ws_size,
                              hipStream_t stream) {
    const float* mu_q    = (const float*)d_in[0];
    const float* sigma_q = (const float*)d_in[1];
    const float* mu_p    = (const float*)d_in[2];
    const float* sigma_p = (const float*)d_in[3];
    float* out      = (float*)d_out;
    float* partials = (float*)d_ws;          // KL_NB * 4 bytes of scratch
    const int n = in_sizes[0];

    _KLD_partials_kernel<<<KL_NB, KL_BLOCK, 0, stream>>>(
        mu_q, sigma_q, mu_p, sigma_p, partials, n);
    _KLD_finalize_kernel<<<1, 32, 0, stream>>>(partials, out, KL_NB);
}